// HandshakingKernel_36060545417529
// MI455X (gfx1250) — compile-verified
//
#include <hip/hip_runtime.h>
#include <hip/hip_bf16.h>
#include <math.h>

// Sizes from the reference
#define BB   8
#define SS   128
#define HH   768
#define PP   8256          // S*(S+1)/2
#define ROWS (BB*SS)       // 1024 unique guide/visible rows
#define TBL  ((size_t)ROWS*HH)

typedef __attribute__((ext_vector_type(2))) float v2f;
typedef __attribute__((ext_vector_type(4))) float v4f;
typedef __attribute__((ext_vector_type(8))) float v8f;

// ---------------------------------------------------------------------------
// Phase 1a: four [1024x768] @ [768x768]^T GEMMs via V_WMMA_F32_16X16X4_F32.
//   gemm 0: T1 = x @ cat_W[:, :768]^T
//   gemm 1: T2 = x @ beta_W^T  + beta
//   gemm 2: T3 = x @ gamma_W^T + gamma
//   gemm 3: T4 = y @ cat_W[:, 768:]^T
// One wave computes one 16x16 output tile; 4 waves per 128-thread block.
// tiles = (1024/16)*(768/16) = 64*48 = 3072 per gemm -> gridDim = (768, 4).
// ---------------------------------------------------------------------------
__global__ __launch_bounds__(128)
void gemm_tables_kernel(const float* __restrict__ x, const float* __restrict__ y,
                        const float* __restrict__ cat_W,
                        const float* __restrict__ beta_W,
                        const float* __restrict__ gamma_W,
                        const float* __restrict__ beta,
                        const float* __restrict__ gamma,
                        float* __restrict__ T1, float* __restrict__ T2,
                        float* __restrict__ T3, float* __restrict__ T4)
{
    const int gemm = blockIdx.y;
    const int lane = threadIdx.x & 31;
    const int wave = threadIdx.x >> 5;
    const int tile = blockIdx.x * 4 + wave;       // 0..3071
    const int tm = tile / 48;                     // 0..63
    const int tn = tile - tm * 48;                // 0..47
    const int m0 = tm * 16;
    const int n0 = tn * 16;

    const int mr  = lane & 15;                    // row (A) / col (B,D) within tile
    const int grp = lane >> 4;                    // 0: K=k,k+1   1: K=k+2,k+3

    const float* X;  const float* W;  int ldw; int hoff;
    const float* bias; float* Out;
    if (gemm == 0)      { X = x; W = cat_W;   ldw = 2*HH; hoff = 0;  bias = nullptr; Out = T1; }
    else if (gemm == 1) { X = x; W = beta_W;  ldw = HH;   hoff = 0;  bias = beta;    Out = T2; }
    else if (gemm == 2) { X = x; W = gamma_W; ldw = HH;   hoff = 0;  bias = gamma;   Out = T3; }
    else                { X = y; W = cat_W;   ldw = 2*HH; hoff = HH; bias = nullptr; Out = T4; }

    // A fragment: A[M=m0+mr][K=k+2*grp .. +1]   (16x4 fp32 layout, ISA 7.12.2)
    // B fragment: B[K][N=n0+mr] = W[n0+mr][hoff+K]  (W row-major => direct v2f load)
    const float* Arow = X + (size_t)(m0 + mr) * HH + grp * 2;
    const float* Brow = W + (size_t)(n0 + mr) * ldw + hoff + grp * 2;

    v8f acc = {};
    #pragma unroll 8
    for (int k = 0; k < HH; k += 4) {
        v2f a = *(const v2f*)(Arow + k);
        v2f b = *(const v2f*)(Brow + k);
        // 8 args: (neg_a, A, neg_b, B, c_mod, C, reuse_a, reuse_b)
        acc = __builtin_amdgcn_wmma_f32_16x16x4_f32(
                  false, a, false, b, (short)0, acc, false, false);
    }

    const float bv = bias ? bias[n0 + mr] : 0.0f;
    // D layout: VGPR v -> M = v + 8*grp, N = mr (ISA 7.12.2, 32-bit C/D 16x16)
    float* Obase = Out + (size_t)(m0 + grp * 8) * HH + n0 + mr;
    #pragma unroll
    for (int v = 0; v < 8; ++v)
        Obase[(size_t)v * HH] = acc[v] + bv;
}

// ---------------------------------------------------------------------------
// Phase 1b: per visible row r: mean/var over H, then
//   T5[r,o] = (y[r,o] - mean) / (var + EPS)^2      (reference squares, no sqrt)
// One 256-thread block per row (1024 rows), 3 elements per thread.
// ---------------------------------------------------------------------------
__global__ __launch_bounds__(256)
void stats_kernel(const float* __restrict__ y, float* __restrict__ T5)
{
    const int row = blockIdx.x;
    const float* yr = y + (size_t)row * HH;
    const int t = threadIdx.x;

    const float v0 = yr[t], v1 = yr[t + 256], v2 = yr[t + 512];

    __shared__ float ss[256];
    __shared__ float sq[256];
    ss[t] = v0 + v1 + v2;
    sq[t] = v0 * v0 + v1 * v1 + v2 * v2;
    __syncthreads();
    #pragma unroll
    for (int off = 128; off > 0; off >>= 1) {
        if (t < off) { ss[t] += ss[t + off]; sq[t] += sq[t + off]; }
        __syncthreads();
    }

    __shared__ float s_mean, s_inv;
    if (t == 0) {
        const float mean = ss[0] * (1.0f / HH);
        float var = sq[0] * (1.0f / HH) - mean * mean;
        var = fmaxf(var, 0.0f);
        const float d = var + 1e-12f;
        s_mean = mean;
        s_inv  = 1.0f / (d * d);        // cen / (var+eps)^2
    }
    __syncthreads();
    const float mean = s_mean, inv = s_inv;

    float* o = T5 + (size_t)row * HH;
    o[t]       = (v0 - mean) * inv;
    o[t + 256] = (v1 - mean) * inv;
    o[t + 512] = (v2 - mean) * inv;
}

// ---------------------------------------------------------------------------
// Phase 2: streaming combine. One block per (b, pair) row; 192 threads,
// one float4 each (768 = 192*4). Tables (15 MB) are L2-resident; the 203 MB
// output is written once with non-temporal B128 stores -> HBM-write bound.
// ---------------------------------------------------------------------------
__global__ __launch_bounds__(192)
void combine_kernel(const float* __restrict__ T1, const float* __restrict__ T2,
                    const float* __restrict__ T3, const float* __restrict__ T4,
                    const float* __restrict__ T5, const float* __restrict__ cat_b,
                    float* __restrict__ out)
{
    const int bp = blockIdx.x;            // 0 .. B*P-1
    const int b  = bp / PP;
    const int p  = bp - b * PP;

    // Decode flat upper-triangular index p -> (i, j), j >= i.
    // start(i) = i*S - i*(i-1)/2 ; i = floor((2S+1 - sqrt((2S+1)^2 - 8p))/2)
    int i = (int)((257.0f - sqrtf(66049.0f - 8.0f * (float)p)) * 0.5f);
    if (i > 127) i = 127;
    if (i < 0)   i = 0;
    int st = i * SS - (i * (i - 1)) / 2;
    while (st > p) { --i; st = i * SS - (i * (i - 1)) / 2; }
    while (i < SS - 1) {
        const int stn = (i + 1) * SS - ((i + 1) * i) / 2;
        if (stn <= p) { ++i; st = stn; } else break;
    }
    const int j = i + (p - st);

    const int o = threadIdx.x * 4;
    const size_t gi = ((size_t)(b * SS + i)) * HH + o;   // guide-row tables
    const size_t gj = ((size_t)(b * SS + j)) * HH + o;   // visible-row tables

    const v4f t1 = *(const v4f*)(T1 + gi);
    const v4f t2 = *(const v4f*)(T2 + gi);
    const v4f t3 = *(const v4f*)(T3 + gi);
    const v4f t4 = *(const v4f*)(T4 + gj);
    const v4f t5 = *(const v4f*)(T5 + gj);
    const v4f cb = *(const v4f*)(cat_b + o);

    v4f cat = t1 + t4 + cb;
    cat.x = fmaxf(cat.x, 0.0f);
    cat.y = fmaxf(cat.y, 0.0f);
    cat.z = fmaxf(cat.z, 0.0f);
    cat.w = fmaxf(cat.w, 0.0f);

    const v4f res = (cat + t5 * t3 + t2) * 0.5f;

    __builtin_nontemporal_store(res, (v4f*)(out + (size_t)bp * HH + o));
}

// ---------------------------------------------------------------------------
extern "C" void kernel_launch(void* const* d_in, const int* in_sizes, int n_in,
                              void* d_out, int out_size, void* d_ws, size_t ws_size,
                              hipStream_t stream)
{
    (void)in_sizes; (void)n_in; (void)out_size; (void)ws_size;
    const float* x       = (const float*)d_in[0];  // (B,S,H)
    const float* y       = (const float*)d_in[1];  // (B,S,H)
    const float* cat_W   = (const float*)d_in[2];  // (H, 2H)
    const float* cat_b   = (const float*)d_in[3];  // (H,)
    const float* beta    = (const float*)d_in[4];  // (H,)
    const float* gamma   = (const float*)d_in[5];  // (H,)
    const float* beta_W  = (const float*)d_in[6];  // (H, H)
    const float* gamma_W = (const float*)d_in[7];  // (H, H)
    float* out = (float*)d_out;

    float* T1 = (float*)d_ws;          // each table: 1024*768 floats (3 MB)
    float* T2 = T1 + TBL;
    float* T3 = T2 + TBL;
    float* T4 = T3 + TBL;
    float* T5 = T4 + TBL;              // total 15 MB of workspace

    gemm_tables_kernel<<<dim3(768, 4), 128, 0, stream>>>(
        x, y, cat_W, beta_W, gamma_W, beta, gamma, T1, T2, T3, T4);

    stats_kernel<<<ROWS, 256, 0, stream>>>(y, T5);

    combine_kernel<<<BB * PP, 192, 0, stream>>>(T1, T2, T3, T4, T5, cat_b, out);
}